// GNN_77421080477944
// MI455X (gfx1250) — compile-verified
//
#include <hip/hip_runtime.h>
#include <math.h>

// ---------------------------------------------------------------------------
// 2-layer GCN (DGL GraphConv, norm='both') for MI455X (gfx1250, wave32).
// Heavy op: X[100000,512] @ W1[512,16] via V_WMMA_F32_16X16X4_F32.
// Edge aggregation via L2-resident gather + global_atomic_add_f32.
// ---------------------------------------------------------------------------

typedef __attribute__((ext_vector_type(2))) float v2f;
typedef __attribute__((ext_vector_type(8))) float v8f;

#define F_IN 512
#define HID  16
#define NCLS 2

// ---------------------------------------------------------------------------
__global__ void gnn_zero(float* __restrict__ ws, int n) {
  int i = blockIdx.x * blockDim.x + threadIdx.x;
  if (i < n) ws[i] = 0.0f;
}

// one thread per edge: degree histograms
__global__ void gnn_degrees(const int* __restrict__ src, const int* __restrict__ dst,
                            float* __restrict__ deg_out, float* __restrict__ deg_in, int E) {
  int e = blockIdx.x * blockDim.x + threadIdx.x;
  if (e < E) {
    atomicAdd(&deg_out[src[e]], 1.0f);
    atomicAdd(&deg_in[dst[e]], 1.0f);
  }
}

__global__ void gnn_invsqrt(const float* __restrict__ deg_out, const float* __restrict__ deg_in,
                            float* __restrict__ inv_out, float* __restrict__ inv_in, int N) {
  int i = blockIdx.x * blockDim.x + threadIdx.x;
  if (i < N) {
    inv_out[i] = rsqrtf(fmaxf(deg_out[i], 1.0f));
    inv_in[i]  = rsqrtf(fmaxf(deg_in[i], 1.0f));
  }
}

// ---------------------------------------------------------------------------
// h1 = (x * inv_out[:,None]) @ W1   via V_WMMA_F32_16X16X4_F32.
// One wave handles a 16-row tile; K=512 swept in steps of 4 (128 WMMAs/tile).
// A frag (lane L): rows M = L&15, K = k + 2*(L>>4) + {0,1}  (contiguous float2)
// B frag (lane L): col  N = L&15, K = k + 2*(L>>4) + {0,1}  (from LDS-staged W1)
// D frag (lane L): elem p -> (M = p + 8*(L>>4), N = L&15)
// ---------------------------------------------------------------------------
__global__ __launch_bounds__(256) void gnn_gemm1_wmma(
    const float* __restrict__ x, const float* __restrict__ W1,
    const float* __restrict__ inv_out, float* __restrict__ h1, int N) {
  __shared__ float wlds[F_IN * HID];  // 32 KB of the 320 KB WGP LDS
  for (int i = threadIdx.x; i < F_IN * HID; i += blockDim.x) wlds[i] = W1[i];
  __syncthreads();

  const int wave = threadIdx.x >> 5;
  const int lane = threadIdx.x & 31;
  const int tiles = N >> 4;                 // N = 100000 -> 6250 exact tiles
  const int tile = blockIdx.x * 8 + wave;
  if (tile >= tiles) return;                // no further barriers below

  const int n16 = lane & 15;                // A-row / B-col / D-col
  const int g   = lane >> 4;                // half-wave K group
  const int row = tile * 16 + n16;
  const float s = inv_out[row];
  const float* __restrict__ xr = x + (size_t)row * F_IN;

  v8f c = {};
  #pragma unroll 4
  for (int k = 0; k < F_IN; k += 4) {
    const int kk = k + 2 * g;
    v2f a = *(const v2f*)(xr + kk);         // 8B-aligned global_load_b64
    a.x *= s;
    a.y *= s;
    v2f b;
    b.x = wlds[(kk + 0) * HID + n16];
    b.y = wlds[(kk + 1) * HID + n16];
    c = __builtin_amdgcn_wmma_f32_16x16x4_f32(
        /*neg_a=*/false, a, /*neg_b=*/false, b,
        /*c_mod=*/(short)0, c, /*reuse_a=*/false, /*reuse_b=*/false);
  }

  #pragma unroll
  for (int p = 0; p < 8; ++p) {
    const int m = tile * 16 + p + 8 * g;
    h1[(size_t)m * HID + n16] = c[p];
  }
}

// ---------------------------------------------------------------------------
// agg1[dst] += h1[src]  : 4 threads per edge, each covers 4 features (float4
// gather hits L2: the h1 table is 6.4 MB and the L2 is 192 MB).
// ---------------------------------------------------------------------------
__global__ void gnn_scatter16(const int* __restrict__ src, const int* __restrict__ dst,
                              const float* __restrict__ h1, float* __restrict__ agg1,
                              long E4) {
  long t = (long)blockIdx.x * blockDim.x + threadIdx.x;
  if (t >= E4) return;
  long e = t >> 2;
  int f = (int)(t & 3) * 4;
  int s = src[e], d = dst[e];
  float4 v = *(const float4*)(h1 + (size_t)s * HID + f);
  float* a = agg1 + (size_t)d * HID + f;
  atomicAdd(a + 0, v.x);
  atomicAdd(a + 1, v.y);
  atomicAdd(a + 2, v.z);
  atomicAdd(a + 3, v.w);
}

// ---------------------------------------------------------------------------
// Node-local layer 2: h = relu(agg1*inv_in + b1); h2 = (h*inv_out) @ W2[16,2]
// ---------------------------------------------------------------------------
__global__ void gnn_layer2(const float* __restrict__ agg1, const float* __restrict__ inv_in,
                           const float* __restrict__ inv_out, const float* __restrict__ b1,
                           const float* __restrict__ W2, float* __restrict__ h2, int N) {
  int i = blockIdx.x * blockDim.x + threadIdx.x;
  if (i >= N) return;
  const float si = inv_in[i];
  const float so = inv_out[i];
  const float* a = agg1 + (size_t)i * HID;
  float acc0 = 0.0f, acc1 = 0.0f;
  #pragma unroll
  for (int k = 0; k < HID; ++k) {
    float hv = fmaxf(fmaf(a[k], si, b1[k]), 0.0f) * so;
    acc0 = fmaf(hv, W2[k * NCLS + 0], acc0);
    acc1 = fmaf(hv, W2[k * NCLS + 1], acc1);
  }
  h2[(size_t)i * NCLS + 0] = acc0;
  h2[(size_t)i * NCLS + 1] = acc1;
}

// agg2[dst] += h2[src] : one thread per edge, 2 features
__global__ void gnn_scatter2(const int* __restrict__ src, const int* __restrict__ dst,
                             const float* __restrict__ h2, float* __restrict__ agg2, int E) {
  int e = blockIdx.x * blockDim.x + threadIdx.x;
  if (e >= E) return;
  int s = src[e], d = dst[e];
  float2 v = *(const float2*)(h2 + (size_t)s * NCLS);
  atomicAdd(&agg2[(size_t)d * NCLS + 0], v.x);
  atomicAdd(&agg2[(size_t)d * NCLS + 1], v.y);
}

// out = log_softmax(agg2*inv_in + b2) over 2 classes
__global__ void gnn_final(const float* __restrict__ agg2, const float* __restrict__ inv_in,
                          const float* __restrict__ b2, float* __restrict__ out, int N) {
  int i = blockIdx.x * blockDim.x + threadIdx.x;
  if (i >= N) return;
  const float si = inv_in[i];
  float v0 = fmaf(agg2[(size_t)i * NCLS + 0], si, b2[0]);
  float v1 = fmaf(agg2[(size_t)i * NCLS + 1], si, b2[1]);
  float m = fmaxf(v0, v1);
  float lse = m + logf(expf(v0 - m) + expf(v1 - m));
  out[(size_t)i * NCLS + 0] = v0 - lse;
  out[(size_t)i * NCLS + 1] = v1 - lse;
}

// ---------------------------------------------------------------------------
extern "C" void kernel_launch(void* const* d_in, const int* in_sizes, int n_in,
                              void* d_out, int out_size, void* d_ws, size_t ws_size,
                              hipStream_t stream) {
  const float* x   = (const float*)d_in[0];
  const int*   src = (const int*)d_in[1];
  const int*   dst = (const int*)d_in[2];
  const float* W1  = (const float*)d_in[3];
  const float* b1  = (const float*)d_in[4];
  const float* W2  = (const float*)d_in[5];
  const float* b2  = (const float*)d_in[6];
  float* out = (float*)d_out;

  const int N = in_sizes[0] / F_IN;   // 100000
  const int E = in_sizes[1];          // 3200000

  // Workspace layout (floats): 40*N total = 16 MB
  float* ws      = (float*)d_ws;
  float* deg_out = ws;                // N
  float* deg_in  = ws + (size_t)N;    // N
  float* inv_out = ws + (size_t)2 * N;
  float* inv_in  = ws + (size_t)3 * N;
  float* h1      = ws + (size_t)4 * N;   // N*16
  float* agg1    = ws + (size_t)20 * N;  // N*16
  float* h2      = ws + (size_t)36 * N;  // N*2
  float* agg2    = ws + (size_t)38 * N;  // N*2

  const int B = 256;
  const int wsFloats = 40 * N;

  gnn_zero<<<(wsFloats + B - 1) / B, B, 0, stream>>>(ws, wsFloats);
  gnn_degrees<<<(E + B - 1) / B, B, 0, stream>>>(src, dst, deg_out, deg_in, E);
  gnn_invsqrt<<<(N + B - 1) / B, B, 0, stream>>>(deg_out, deg_in, inv_out, inv_in, N);

  const int tiles = N >> 4;                       // 6250 16-row tiles
  gnn_gemm1_wmma<<<(tiles + 7) / 8, B, 0, stream>>>(x, W1, inv_out, h1, N);

  const long E4 = (long)E * 4;
  gnn_scatter16<<<(int)((E4 + B - 1) / B), B, 0, stream>>>(src, dst, h1, agg1, E4);
  gnn_layer2<<<(N + B - 1) / B, B, 0, stream>>>(agg1, inv_in, inv_out, b1, W2, h2, N);
  gnn_scatter2<<<(E + B - 1) / B, B, 0, stream>>>(src, dst, h2, agg2, E);
  gnn_final<<<(N + B - 1) / B, B, 0, stream>>>(agg2, inv_in, b2, out, N);
}